// BehaviorVelocity_43748536877194
// MI455X (gfx1250) — compile-verified
//
#include <hip/hip_runtime.h>
#include <cstdint>

#define HW_   (512*512)     // 2^18
#define W_    512
#define CH_   20            // NUM_ELEMENTS + 6
#define NB_   4

typedef __attribute__((ext_vector_type(2))) float v2f;
typedef __attribute__((ext_vector_type(8))) float v8f;

// ---------------------------------------------------------------------------
// Kernel 1: per-pixel direction bucket (folds vel_enough), + vdelta init copy.
// dircode = bucket(0..7) if |v| > 0.1 else 255.
// ---------------------------------------------------------------------------
__global__ __launch_bounds__(256) void k_dircode(const float* __restrict__ vel,
                                                 float* __restrict__ vdelta,
                                                 uint8_t* __restrict__ dc) {
    int idx = blockIdx.x * 256 + threadIdx.x;   // 0 .. NB*HW-1
    int b   = idx >> 18;
    int pix = idx & (HW_ - 1);
    float v0 = vel[(size_t)(b*2 + 0)*HW_ + pix];
    float v1 = vel[(size_t)(b*2 + 1)*HW_ + pix];
    vdelta[(size_t)(b*2 + 0)*HW_ + pix] = v0;   // vdelta starts equal to vel
    vdelta[(size_t)(b*2 + 1)*HW_ + pix] = v1;
    float mag = sqrtf(v0*v0 + v1*v1);
    float raw = 0.15915494309189535f * acosf(v1 / (mag + 0.001f));
    float ang = (v0 < 0.0f) ? (1.0f - raw) : raw;
    float bucket = fmodf(floorf(ang * 8.0f + 0.5f), 8.0f);
    dc[idx] = (mag > 0.1f) ? (uint8_t)(int)bucket : (uint8_t)255;
}

// ---------------------------------------------------------------------------
// Aligned-float4 row read with compile-time shift S in {-1,0,+1}:
// out[i] = row[x0 + i + S], zero outside [0,512).
// ---------------------------------------------------------------------------
template<int S>
__device__ __forceinline__ void load_shift(const float* __restrict__ row,
                                           int x0, float o[4]) {
    const float4 f = *reinterpret_cast<const float4*>(row + x0);
    if (S == 0) {
        o[0] = f.x; o[1] = f.y; o[2] = f.z; o[3] = f.w;
    } else if (S == 1) {
        o[0] = f.y; o[1] = f.z; o[2] = f.w;
        o[3] = (x0 + 4 < 512) ? row[x0 + 4] : 0.0f;
    } else {
        o[1] = f.x; o[2] = f.y; o[3] = f.z;
        o[0] = (x0 - 1 >= 0) ? row[x0 - 1] : 0.0f;
    }
}

// ---------------------------------------------------------------------------
// Kernel 2: one direction step of the CA, 4 pixels per lane (B128 traffic).
//   sa[p] = (dir[p]==a) & empty(p+d) & !wall(p);  sb[p] = sa[p-d]
//   w'[c,p] = (1-sa-sb)*w[c,p] + sa*empty_vec[c] + sb*w[c,p-d]
//   vdelta[p] += -0.5*sa*vel[p] + 0.5*sb*vel[p-d]
// ---------------------------------------------------------------------------
template<int DY, int DX>
__global__ __launch_bounds__(256) void k_dir_step(
        const float* __restrict__ src, float* __restrict__ dst,
        const uint8_t* __restrict__ dc, const float* __restrict__ vel,
        float* __restrict__ vdelta, const float* __restrict__ elemEmpty,
        int a) {
    int tid  = blockIdx.x * 256 + threadIdx.x;   // 0 .. NB*HW/4-1
    int idx4 = tid << 2;
    int b    = idx4 >> 18;
    int pix  = idx4 & (HW_ - 1);
    int y    = pix >> 9;
    int x0   = pix & 511;                        // multiple of 4
    size_t wb = (size_t)b * CH_ * HW_;

    int  ya    = y + DY;
    bool in_ya = (unsigned)ya < 512u;
    int  qy    = y - DY;
    bool in_qy = (unsigned)qy < 512u;

    // --- mask inputs ---
    float w0p[4], w1p[4];
    load_shift<0>(src + wb + ((size_t)0*HW_) + (y << 9), x0, w0p);  // EMPTY at p
    load_shift<0>(src + wb + ((size_t)1*HW_) + (y << 9), x0, w1p);  // WALL  at p

    float ea[4] = {0.f, 0.f, 0.f, 0.f};          // EMPTY at p+d (zero OOB)
    if (in_ya) load_shift<DX>(src + wb + ((ya & 511) << 9), x0, ea);

    float wq1[4] = {0.f, 0.f, 0.f, 0.f};         // WALL at q (zero OOB)
    if (in_qy) load_shift<-DX>(src + wb + HW_ + ((qy & 511) << 9), x0, wq1);

    uint32_t dcp = *reinterpret_cast<const uint32_t*>(dc + (b << 18) + pix);
    int dcq[4];
    #pragma unroll
    for (int i = 0; i < 4; ++i) {
        int xq = x0 + i - DX;
        bool ok = in_qy && ((unsigned)xq < 512u);
        dcq[i] = ok ? (int)dc[(b << 18) + ((qy & 511) << 9) + (xq & 511)] : 255;
    }

    float saf[4], sbf[4], coef[4];
    bool any = false;
    #pragma unroll
    for (int i = 0; i < 4; ++i) {
        int  dp = (int)((dcp >> (8*i)) & 255u);
        bool sa = (dp == a) && in_ya && (ea[i] > 0.5f) && !(w1p[i] > 0.5f);
        bool sb = (dcq[i] == a) && (w0p[i] > 0.5f) && !(wq1[i] > 0.5f);
        saf[i]  = sa ? 1.0f : 0.0f;
        sbf[i]  = sb ? 1.0f : 0.0f;
        coef[i] = 1.0f - saf[i] - sbf[i];
        any = any || sa || sb;
    }

    // --- 20-channel update, B128 in / B128 out ---
    #pragma unroll 5
    for (int c = 0; c < CH_; ++c) {
        const float* rp = src + wb + (size_t)c*HW_ + (y << 9);
        float4 wp = *reinterpret_cast<const float4*>(rp + x0);
        float wq[4] = {0.f, 0.f, 0.f, 0.f};
        if (in_qy) load_shift<-DX>(src + wb + (size_t)c*HW_ + ((qy & 511) << 9), x0, wq);
        float ev = elemEmpty[c];
        float4 o;
        o.x = coef[0]*wp.x + saf[0]*ev + sbf[0]*wq[0];
        o.y = coef[1]*wp.y + saf[1]*ev + sbf[1]*wq[1];
        o.z = coef[2]*wp.z + saf[2]*ev + sbf[2]*wq[2];
        o.w = coef[3]*wp.w + saf[3]*ev + sbf[3]*wq[3];
        *reinterpret_cast<float4*>(dst + wb + (size_t)c*HW_ + (y << 9) + x0) = o;
    }

    // --- velocity delta accumulation (sparse) ---
    if (any) {
        size_t vb = (size_t)b * 2 * HW_;
        #pragma unroll
        for (int comp = 0; comp < 2; ++comp) {
            const float* vr = vel + vb + (size_t)comp*HW_ + (y << 9);
            float4 vp = *reinterpret_cast<const float4*>(vr + x0);
            float vq[4] = {0.f, 0.f, 0.f, 0.f};
            if (in_qy) load_shift<-DX>(vel + vb + (size_t)comp*HW_ + ((qy & 511) << 9), x0, vq);
            float* dr = vdelta + vb + (size_t)comp*HW_ + (y << 9);
            float4 dv = *reinterpret_cast<float4*>(dr + x0);
            dv.x += -saf[0]*0.5f*vp.x + sbf[0]*0.5f*vq[0];
            dv.y += -saf[1]*0.5f*vp.y + sbf[1]*0.5f*vq[1];
            dv.z += -saf[2]*0.5f*vp.z + sbf[2]*0.5f*vq[2];
            dv.w += -saf[3]*0.5f*vp.w + sbf[3]*0.5f*vq[3];
            *reinterpret_cast<float4*>(dr + x0) = dv;
        }
    }
}

// ---------------------------------------------------------------------------
// Kernel 3: vel *= 0.95; smooth = conv3x3(ones/18) + 0.5*self; write vel out
// and world channels 18..19. One 16x16 tile per wave32. 18x18 halo tile staged
// into LDS via async global->LDS loads; vertical 3-tap as Y = S(tridiag)*X via
// 4 chained v_wmma_f32_16x16x4_f32; horizontal 3-tap via wave32 shuffles.
// ---------------------------------------------------------------------------
__global__ __launch_bounds__(256) void k_smooth(const float* __restrict__ velB,
                                                float* __restrict__ velOut,
                                                float* __restrict__ worldOut) {
    __shared__ float lds[8][18*20];            // 8 waves * 18 rows * 20 cols
    int lane = threadIdx.x & 31;
    int warp = threadIdx.x >> 5;
    int t  = blockIdx.x * 8 + warp;            // tile id, 0..8191
    int tx = t & 31;
    int ty = (t >> 5) & 31;
    int ch = (t >> 10) & 1;
    int b  = t >> 11;
    float* tile = &lds[warp][0];
    const float* sp = velB + (size_t)(b*2 + ch) * HW_;
    uint64_t sbase = (uint64_t)(uintptr_t)sp;
    int r0 = ty * 16, c0 = tx * 16;

    // Stage 18x18 halo tile into LDS (zero-padded at image borders).
    for (int i = lane; i < 324; i += 32) {
        int r  = i / 18;
        int cc = i - r * 18;
        int gy = r0 - 1 + r, gx = c0 - 1 + cc;
        if (((unsigned)gy < 512u) && ((unsigned)gx < 512u)) {
            uint32_t loff = (uint32_t)(uintptr_t)&tile[r*20 + cc];
            int goff = (((gy) << 9) | gx) * 4;
            asm volatile("global_load_async_to_lds_b32 %0, %1, %2"
                         :: "v"(loff), "v"(goff), "s"(sbase) : "memory");
        } else {
            tile[r*20 + cc] = 0.0f;
        }
    }
    asm volatile("s_wait_asynccnt 0\n\ts_wait_dscnt 0" ::: "memory");
    __builtin_amdgcn_wave_barrier();

    int hi = lane >> 4;        // half-wave select
    int n  = lane & 15;        // output column within tile
    int m  = n;                // A-matrix row = lane&15 (ISA layout)
    v8f acc = {0.f,0.f,0.f,0.f,0.f,0.f,0.f,0.f};

#if __has_builtin(__builtin_amdgcn_wmma_f32_16x16x4_f32)
    // Vertical pass: Y = S * X, S[m,k] = (|m-k|<=1).  A 16x4 chunks: lane m=l&15,
    // VGPR0 holds K=4k+2*hi, VGPR1 holds K=4k+2*hi+1 (ISA 32-bit A layout);
    // B 4x16 chunks mirror the same (0,2)/(1,3) half-wave split.
    #pragma unroll
    for (int k = 0; k < 4; ++k) {
        int K0 = 4*k + 2*hi;
        int K1 = K0 + 1;
        v2f av, bv;
        av.x = (m - K0 <= 1 && K0 - m <= 1) ? 1.0f : 0.0f;
        av.y = (m - K1 <= 1 && K1 - m <= 1) ? 1.0f : 0.0f;
        bv.x = tile[(1 + K0)*20 + 1 + n];
        bv.y = tile[(1 + K1)*20 + 1 + n];
        acc = __builtin_amdgcn_wmma_f32_16x16x4_f32(false, av, false, bv,
                                                    (short)0, acc, false, false);
    }
    // Halo rows: Y[0,:] += X[-1,:], Y[15,:] += X[16,:].
    // Constant element indices + predicated value (avoids dynamic v8f indexing).
    {
        float topv = tile[0*20  + 1 + n];
        float botv = tile[17*20 + 1 + n];
        acc[0] += (hi == 0) ? topv : 0.0f;
        acc[7] += (hi == 1) ? botv : 0.0f;
    }
#else
    // VALU fallback: direct vertical 3-tap including halo rows.
    #pragma unroll
    for (int v = 0; v < 8; ++v) {
        int row = v + 8*hi;
        acc[v] = tile[row*20 + 1 + n] + tile[(row+1)*20 + 1 + n]
               + tile[(row+2)*20 + 1 + n];
    }
#endif

    // Horizontal 3-tap via wave32 shuffles on C layout (VGPR v: row v+8*hi, col n)
    const float inv18 = 1.0f / 18.0f;
    #pragma unroll
    for (int v = 0; v < 8; ++v) {
        int row = v + 8*hi;
        float cur = acc[v];
        float lft = __shfl(cur, lane - 1, 32);
        float rgt = __shfl(cur, lane + 1, 32);
        if (n == 0)
            lft = tile[row*20 + 0]  + tile[(row+1)*20 + 0]  + tile[(row+2)*20 + 0];
        if (n == 15)
            rgt = tile[row*20 + 17] + tile[(row+1)*20 + 17] + tile[(row+2)*20 + 17];
        float center = tile[(row+1)*20 + 1 + n];
        float out = 0.95f * ((lft + cur + rgt) * inv18 + 0.5f * center);
        int gy = r0 + row, gx = c0 + n;
        size_t o = (size_t)(b*2 + ch)*HW_ + ((gy << 9) | gx);
        velOut[o] = out;
        worldOut[(size_t)(b*CH_ + 18 + ch)*HW_ + ((gy << 9) | gx)] = out;
    }
}

// ---------------------------------------------------------------------------
static void run_step(const float* src, float* dst, const uint8_t* dc,
                     const float* vel, float* vdelta, const float* elemEmpty,
                     int a, hipStream_t stream) {
    dim3 blk(256), grd(NB_*HW_/4/256);
#define LAUNCH_STEP(DYv, DXv) \
    k_dir_step<DYv, DXv><<<grd, blk, 0, stream>>>(src, dst, dc, vel, vdelta, elemEmpty, a)
    switch (a) {
        case 0: LAUNCH_STEP(-1, 0); break;
        case 1: LAUNCH_STEP(-1, 1); break;
        case 2: LAUNCH_STEP( 0, 1); break;
        case 3: LAUNCH_STEP( 1, 1); break;
        case 4: LAUNCH_STEP( 1, 0); break;
        case 5: LAUNCH_STEP( 1,-1); break;
        case 6: LAUNCH_STEP( 0,-1); break;
        case 7: LAUNCH_STEP(-1,-1); break;
    }
#undef LAUNCH_STEP
}

extern "C" void kernel_launch(void* const* d_in, const int* in_sizes, int n_in,
                              void* d_out, int out_size, void* d_ws, size_t ws_size,
                              hipStream_t stream) {
    (void)in_sizes; (void)n_in; (void)out_size; (void)ws_size;
    const float* worldIn   = (const float*)d_in[0];   // 4*20*512*512
    const float* velIn     = (const float*)d_in[1];   // 4*2*512*512
    const float* elemEmpty = (const float*)d_in[2];   // 20

    float* worldOut = (float*)d_out;                          // world region
    float* velOut   = worldOut + (size_t)NB_*CH_*HW_;         // vel region

    float*  worldA = (float*)d_ws;                            // 80 MB ping buffer
    float*  velA   = worldA + (size_t)NB_*CH_*HW_;            // 8 MB
    float*  velB   = velA   + (size_t)NB_*2*HW_;              // 8 MB
    uint8_t* dc    = (uint8_t*)(velB + (size_t)NB_*2*HW_);    // 1 MB

    dim3 blk(256), grdP(NB_*HW_/256);

    float* dsts[2] = {worldA, worldOut};   // s even -> worldA, s odd -> worldOut
    const float* src = worldIn;
    int s = 0;

    // ---- iteration 1 ----
    k_dircode<<<grdP, blk, 0, stream>>>(velIn, velA, dc);
    for (int a = 0; a < 8; ++a, ++s) {
        float* dst = dsts[s & 1];
        run_step(src, dst, dc, velIn, velA, elemEmpty, a, stream);
        src = dst;
    }
    // ---- iteration 2 ----
    k_dircode<<<grdP, blk, 0, stream>>>(velA, velB, dc);
    for (int a = 0; a < 8; ++a, ++s) {
        float* dst = dsts[s & 1];
        run_step(src, dst, dc, velA, velB, elemEmpty, a, stream);
        src = dst;
    }
    // s == 16; last write (s=15, odd) landed in worldOut.

    // ---- scale + smooth velocity, write vel out + world ch 18..19 ----
    k_smooth<<<dim3(8192/8), blk, 0, stream>>>(velB, velOut, worldOut);
}